// OCGatherEnergyCorrFac_61237643706562
// MI455X (gfx1250) — compile-verified
//
#include <hip/hip_runtime.h>
#include <hip/hip_bf16.h>

// Workspace layout (floats), S = N_SHOWERS+1 = 10001:
//   ws[0   ..  S) : e_hit_shower
//   ws[S   .. 2S) : e_track_shower
//   ws[2S  .. 3S) : e_hit_shower_corr
//   ws[3S  .. 4S) : e_track_shower_corr
// Total 4*10001*4 B ~= 160 KB of d_ws.

typedef __attribute__((ext_vector_type(2))) float v2f;
typedef __attribute__((ext_vector_type(8))) float v8f;

// ---------------------------------------------------------------------------
// Pass 0: zero the shower accumulators (d_ws is poisoned by the harness).
// ---------------------------------------------------------------------------
__global__ void zero_kernel(float* __restrict__ p, int n) {
  int i = blockIdx.x * blockDim.x + threadIdx.x;
  if (i < n) p[i] = 0.0f;
}

// ---------------------------------------------------------------------------
// Pass 1: segment-sum per-hit energies into shower bins.
// Exactly one global f32 atomic per non-noise hit (noise adds 0 -> skipped).
// ---------------------------------------------------------------------------
__device__ __forceinline__ void accum_one(int s, float e, int id,
                                          float* __restrict__ eh,
                                          float* __restrict__ et) {
  if (s >= 0) {                       // noise (sid==-1) contributes nothing
    float* dst = (id == 0) ? eh : et; // recHitID in {0,1}
    atomicAdd(dst + (s + 1), e);
  }
}

__global__ void accumulate_kernel(const int* __restrict__ sid,
                                  const float* __restrict__ energy,
                                  const int* __restrict__ rid,
                                  float* __restrict__ eh,
                                  float* __restrict__ et, int n) {
  const int stride = gridDim.x * blockDim.x;
  const int tid = blockIdx.x * blockDim.x + threadIdx.x;
  const int nq = n >> 2;
  const int4*   sid4 = (const int4*)sid;
  const float4* e4   = (const float4*)energy;
  const int4*   id4  = (const int4*)rid;
  for (int q = tid; q < nq; q += stride) {
    // CDNA5 speculative prefetch (global_prefetch_b8); OOB is silently dropped
    __builtin_prefetch(&sid4[q + stride], 0, 0);
    __builtin_prefetch(&e4[q + stride], 0, 0);
    __builtin_prefetch(&id4[q + stride], 0, 0);
    int4 s = sid4[q];
    float4 e = e4[q];
    int4 d = id4[q];
    accum_one(s.x, e.x, d.x, eh, et);
    accum_one(s.y, e.y, d.y, eh, et);
    accum_one(s.z, e.z, d.z, eh, et);
    accum_one(s.w, e.w, d.w, eh, et);
  }
  for (int j = (nq << 2) + tid; j < n; j += stride)  // scalar tail (n%4)
    accum_one(sid[j], energy[j], rid[j], eh, et);
}

// ---------------------------------------------------------------------------
// Pass 2: per-shower corrected energies via V_WMMA_F32_16X16X4_F32.
// For a 16-shower tile: A has the shower energies in column K=0, B has the
// correction factors in row K=0 -> D = outer product; diag(D)[m] = e[m]*c[m],
// bit-exact with a scalar f32 multiply (single product, +0.0 accumulate, RNE).
// Diagonal is extracted through a private per-wave LDS slab; LDS ops from one
// wave complete in order, s_wait_dscnt 0 + memory clobber pin the ordering.
// ---------------------------------------------------------------------------
__device__ __forceinline__ float load_corr(int s, const int* __restrict__ alpha,
                                           const int* __restrict__ sid,
                                           const float* __restrict__ pcf,
                                           int n_hits) {
  if (s == 0) return 0.0f;            // prepended zero (segment 0 = noise bin)
  int a = alpha[s - 1];               // in [0, n_hits]
  if (a >= n_hits) return 0.0f;       // pcf_ext zero pad
  return (sid[a] == -1) ? 0.0f : pcf[a];
}

__global__ void shower_corr_wmma_kernel(const int* __restrict__ pred_sid,
                                        const float* __restrict__ pcf,
                                        const int* __restrict__ alpha_tracks,
                                        const int* __restrict__ alpha_hits,
                                        const float* __restrict__ eh_sh,
                                        const float* __restrict__ et_sh,
                                        float* __restrict__ ehc_sh,
                                        float* __restrict__ etc_sh,
                                        int n_seg, int n_hits) {
  __shared__ float tile_h[8][256];    // one 16x16 f32 D-tile per wave
  __shared__ float tile_t[8][256];
  const int lane   = threadIdx.x & 31;
  const int wIdx   = threadIdx.x >> 5;                         // wave in block
  const int wave   = (blockIdx.x * blockDim.x + threadIdx.x) >> 5;
  const int nwaves = (gridDim.x * blockDim.x) >> 5;
  const int ntiles = (n_seg + 15) >> 4;
  const int col    = lane & 15;
  const int half   = lane >> 4;

  for (int t = wave; t < ntiles; t += nwaves) {   // wave-uniform bound: EXEC
    int s = (t << 4) + col;                       // stays all-ones for WMMA
    const bool valid = (half == 0) && (s < n_seg);
    float xh = 0.0f, xt = 0.0f, yh = 0.0f, yt = 0.0f;
    if (valid) {
      xh = eh_sh[s];
      xt = et_sh[s];
      yh = load_corr(s, alpha_hits, pred_sid, pcf, n_hits);
      yt = load_corr(s, alpha_tracks, pred_sid, pcf, n_hits);
    }
    // A(16x4): VGPR0 lanes0-15 = (M=lane,K=0); everything else zero.
    // B(4x16): VGPR0 lanes0-15 = (K=0,N=lane); everything else zero.
    v2f ah = {xh, 0.0f}, bh = {yh, 0.0f};
    v2f at = {xt, 0.0f}, bt = {yt, 0.0f};
    v8f c0 = {};
#if __has_builtin(__builtin_amdgcn_wmma_f32_16x16x4_f32)
    v8f dh = __builtin_amdgcn_wmma_f32_16x16x4_f32(false, ah, false, bh,
                                                   (short)0, c0, false, false);
    v8f dt = __builtin_amdgcn_wmma_f32_16x16x4_f32(false, at, false, bt,
                                                   (short)0, c0, false, false);
#else
    v8f dh = c0, dt = c0;             // scalar fallback: diag only
    dh[0] = xh * yh;                  // (read directly as dh[0] below, no LDS)
    dt[0] = xt * yt;
#endif
#if __has_builtin(__builtin_amdgcn_wmma_f32_16x16x4_f32)
    // D(16x16) f32 layout: VGPR r -> M=r (lanes 0-15) / M=r+8 (lanes 16-31),
    // N = lane%16. Scatter to LDS as tile[M*16+N], then read the diagonal.
#pragma unroll
    for (int r = 0; r < 8; ++r) {
      int row = r + (half << 3);
      tile_h[wIdx][row * 16 + col] = dh[r];
      tile_t[wIdx][row * 16 + col] = dt[r];
    }
    asm volatile("s_wait_dscnt 0" ::: "memory");  // intra-wave LDS RAW fence
    float vh = tile_h[wIdx][col * 17];            // element (col, col)
    float vt = tile_t[wIdx][col * 17];
#else
    float vh = dh[0];
    float vt = dt[0];
#endif
    if (valid) {
      ehc_sh[s] = vh;
      etc_sh[s] = vt;
    }
  }
}

// ---------------------------------------------------------------------------
// Pass 3: per-hit gather of the 4 shower quantities, coalesced float4 stores.
// Output order: e_track_raw, e_track_corrected, e_hit_raw, e_hit_corrected.
// Shower tables (160 KB) are L2-resident on MI455X (192 MB L2).
// ---------------------------------------------------------------------------
__global__ void gather_out_kernel(const int* __restrict__ sid,
                                  const float* __restrict__ et_sh,
                                  const float* __restrict__ etc_sh,
                                  const float* __restrict__ eh_sh,
                                  const float* __restrict__ ehc_sh,
                                  float* __restrict__ out, int n) {
  const int stride = gridDim.x * blockDim.x;
  const int tid = blockIdx.x * blockDim.x + threadIdx.x;
  const int nq = n >> 2;
  const int4* sid4 = (const int4*)sid;
  float4* o_tr = (float4*)(out);
  float4* o_tc = (float4*)(out + (size_t)n);
  float4* o_hr = (float4*)(out + (size_t)2 * n);
  float4* o_hc = (float4*)(out + (size_t)3 * n);
  for (int q = tid; q < nq; q += stride) {
    __builtin_prefetch(&sid4[q + stride], 0, 0);
    int4 s = sid4[q];
    int ax = s.x + 1, ay = s.y + 1, az = s.z + 1, aw = s.w + 1;
    o_tr[q] = make_float4(et_sh[ax],  et_sh[ay],  et_sh[az],  et_sh[aw]);
    o_tc[q] = make_float4(etc_sh[ax], etc_sh[ay], etc_sh[az], etc_sh[aw]);
    o_hr[q] = make_float4(eh_sh[ax],  eh_sh[ay],  eh_sh[az],  eh_sh[aw]);
    o_hc[q] = make_float4(ehc_sh[ax], ehc_sh[ay], ehc_sh[az], ehc_sh[aw]);
  }
  for (int j = (nq << 2) + tid; j < n; j += stride) {  // scalar tail
    int a = sid[j] + 1;
    out[j]                 = et_sh[a];
    out[(size_t)n + j]     = etc_sh[a];
    out[(size_t)2 * n + j] = eh_sh[a];
    out[(size_t)3 * n + j] = ehc_sh[a];
  }
}

// ---------------------------------------------------------------------------
extern "C" void kernel_launch(void* const* d_in, const int* in_sizes, int n_in,
                              void* d_out, int out_size, void* d_ws,
                              size_t ws_size, hipStream_t stream) {
  // setup_inputs() order:
  const int*   pred_sid     = (const int*)d_in[0];
  const float* pcf          = (const float*)d_in[1];   // pred_energy_corr_factor
  const float* energy       = (const float*)d_in[2];   // recHitEnergy
  const int*   rid          = (const int*)d_in[3];     // recHitID
  // d_in[4] = pred_beta: unused by the reference outputs
  const int*   alpha_tracks = (const int*)d_in[5];
  const int*   alpha_hits   = (const int*)d_in[6];

  const int n = in_sizes[0];        // N_HITS
  const int S = in_sizes[5] + 1;    // N_SHOWERS + 1 segments

  float* ws     = (float*)d_ws;
  float* eh_sh  = ws;
  float* et_sh  = ws + S;
  float* ehc_sh = ws + 2 * S;
  float* etc_sh = ws + 3 * S;
  float* out    = (float*)d_out;

  zero_kernel<<<(2 * S + 255) / 256, 256, 0, stream>>>(ws, 2 * S);
  accumulate_kernel<<<2048, 256, 0, stream>>>(pred_sid, energy, rid,
                                              eh_sh, et_sh, n);
  shower_corr_wmma_kernel<<<20, 256, 0, stream>>>(pred_sid, pcf, alpha_tracks,
                                                  alpha_hits, eh_sh, et_sh,
                                                  ehc_sh, etc_sh, S, n);
  gather_out_kernel<<<2048, 256, 0, stream>>>(pred_sid, et_sh, etc_sh,
                                              eh_sh, ehc_sh, out, n);
}